// OtsuThresholdLayer_8873402433666
// MI455X (gfx1250) — compile-verified
//
#include <hip/hip_runtime.h>
#include <stdint.h>

// ---------------- problem constants (x: [64, 512, 512, 3] f32) ----------------
#define NIMG   64
#define NPIX   (512 * 512)          // pixels per image = 262144
#define NBINS  256
#define BLK    256
#define GROUPS_PER_IMG (NPIX / 4)   // each thread handles 4 pixels -> 65536 groups
#define HIST_ITERS 4                // hist kernel: 4 groups per thread

typedef float v4f __attribute__((ext_vector_type(4)));
typedef int   v4i __attribute__((vector_size(4 * sizeof(int))));   // matches builtin param type

// CDNA5 async global->LDS path, probed via __has_builtin (clean fallback otherwise)
#if defined(__gfx1250__) && __has_builtin(__builtin_amdgcn_global_load_async_to_lds_b128)
#define HAS_ASYNC_LDS 1
typedef __attribute__((address_space(1))) v4i* async_gptr_t;   // global (device) int4*
typedef __attribute__((address_space(3))) v4i* async_lptr_t;   // LDS (shared) int4*
#else
#define HAS_ASYNC_LDS 0
#endif

template <int N>
__device__ __forceinline__ void wait_async() {
#if defined(__gfx1250__)
#if __has_builtin(__builtin_amdgcn_s_wait_asynccnt)
    __builtin_amdgcn_s_wait_asynccnt(N);
#else
    asm volatile("s_wait_asynccnt %0" :: "i"(N) : "memory");
#endif
#endif
}

// gray = 0.2989 R + 0.5870 G + 0.1140 B, 4 pixels per thread (3 NT float4 loads)
__device__ __forceinline__ v4f gray4_from_x(const float* __restrict__ x, long long pix) {
    const v4f* xv = (const v4f*)(x + pix * 3);
    v4f a = __builtin_nontemporal_load(xv + 0);
    v4f b = __builtin_nontemporal_load(xv + 1);
    v4f c = __builtin_nontemporal_load(xv + 2);
    v4f r;
    r.x = fmaf(0.2989f, a.x, fmaf(0.5870f, a.y, 0.1140f * a.z));
    r.y = fmaf(0.2989f, a.w, fmaf(0.5870f, b.x, 0.1140f * b.y));
    r.z = fmaf(0.2989f, b.z, fmaf(0.5870f, b.w, 0.1140f * c.x));
    r.w = fmaf(0.2989f, c.y, fmaf(0.5870f, c.z, 0.1140f * c.w));
    return r;
}

__device__ __forceinline__ void histo4(v4f gv, float mn, float scale, unsigned* sh) {
    int b0 = min(max((int)((gv.x - mn) * scale), 0), NBINS - 1);
    int b1 = min(max((int)((gv.y - mn) * scale), 0), NBINS - 1);
    int b2 = min(max((int)((gv.z - mn) * scale), 0), NBINS - 1);
    int b3 = min(max((int)((gv.w - mn) * scale), 0), NBINS - 1);
    atomicAdd(&sh[b0], 1u);
    atomicAdd(&sh[b1], 1u);
    atomicAdd(&sh[b2], 1u);
    atomicAdd(&sh[b3], 1u);
}

// ---------------- pass 0: init scratch ----------------
__global__ __launch_bounds__(BLK) void otsu_init_k(unsigned* __restrict__ hist,
                                                   unsigned* __restrict__ minmax) {
    int i = blockIdx.x * BLK + threadIdx.x;
    if (i < NIMG * NBINS) {
        hist[i] = 0u;
    } else if (i < NIMG * NBINS + NIMG) {
        minmax[2 * (i - NIMG * NBINS) + 0] = 0x7F7FFFFFu;  // +FLT_MAX bits (min seed)
    } else if (i < NIMG * NBINS + 2 * NIMG) {
        minmax[2 * (i - NIMG * NBINS - NIMG) + 1] = 0u;    // 0.0f bits (max seed; gray >= 0)
    }
}

// ---------------- pass 1: grayscale + per-image min/max ----------------
__global__ __launch_bounds__(BLK) void otsu_gray_minmax_k(const float* __restrict__ x,
                                                          float* __restrict__ gray,
                                                          unsigned* __restrict__ minmax) {
    const int b = blockIdx.y;
    const int g = blockIdx.x * BLK + threadIdx.x;                  // 0..65535
    const long long pix = (long long)b * NPIX + (long long)g * 4;
    v4f gv = gray4_from_x(x, pix);
    if (gray) *(v4f*)(gray + pix) = gv;                            // temporal: stays in 192MB L2

    float mn = fminf(fminf(gv.x, gv.y), fminf(gv.z, gv.w));
    float mx = fmaxf(fmaxf(gv.x, gv.y), fmaxf(gv.z, gv.w));
    // wave32 shuffle reduction
    #pragma unroll
    for (int off = 16; off > 0; off >>= 1) {
        mn = fminf(mn, __shfl_xor(mn, off, 32));
        mx = fmaxf(mx, __shfl_xor(mx, off, 32));
    }
    __shared__ float smn[BLK / 32], smx[BLK / 32];
    const int wave = threadIdx.x >> 5, lane = threadIdx.x & 31;
    if (lane == 0) { smn[wave] = mn; smx[wave] = mx; }
    __syncthreads();
    if (threadIdx.x == 0) {
        float m = smn[0], M = smx[0];
        #pragma unroll
        for (int i = 1; i < BLK / 32; ++i) { m = fminf(m, smn[i]); M = fmaxf(M, smx[i]); }
        atomicMin(&minmax[2 * b + 0], __float_as_uint(m));   // valid ordering: gray >= 0
        atomicMax(&minmax[2 * b + 1], __float_as_uint(M));
    }
}

// ---------------- pass 2: per-image 256-bin histogram ----------------
__global__ __launch_bounds__(BLK) void otsu_hist_k(const float* __restrict__ x,
                                                   const float* __restrict__ gray,
                                                   const unsigned* __restrict__ minmax,
                                                   unsigned* __restrict__ hist) {
    const int b = blockIdx.y;
    __shared__ unsigned sh[NBINS];
#if HAS_ASYNC_LDS
    __shared__ v4f tile[HIST_ITERS][BLK];
#endif
    sh[threadIdx.x] = 0u;
    __syncthreads();

    const float mn = __uint_as_float(minmax[2 * b + 0]);
    const float mx = __uint_as_float(minmax[2 * b + 1]);
    const float scale = 256.0f / fmaxf(mx - mn, 1e-12f);

    if (gray) {
#if HAS_ASYNC_LDS
        // Pipelined gfx1250 async copies: issue ALL tiles, then drain with descending
        // s_wait_asynccnt (async loads complete in order), keeping loads in flight.
        const long long pixBase =
            (long long)b * NPIX +
            (long long)(blockIdx.x * (BLK * HIST_ITERS) + threadIdx.x) * 4;
        #pragma unroll
        for (int i = 0; i < HIST_ITERS; ++i) {
            __builtin_amdgcn_global_load_async_to_lds_b128(
                (async_gptr_t)(gray + pixBase + (long long)i * (BLK * 4)),
                (async_lptr_t)&tile[i][threadIdx.x],
                0, 0);
        }
#define PROCESS_TILE(i, rem)                                  \
        do {                                                  \
            wait_async<rem>();                                \
            asm volatile("" ::: "memory");                    \
            v4f gv = tile[i][threadIdx.x];                    \
            histo4(gv, mn, scale, sh);                        \
        } while (0)
        PROCESS_TILE(0, 3);
        PROCESS_TILE(1, 2);
        PROCESS_TILE(2, 1);
        PROCESS_TILE(3, 0);
#undef PROCESS_TILE
#else
        #pragma unroll
        for (int i = 0; i < HIST_ITERS; ++i) {
            const int g = blockIdx.x * (BLK * HIST_ITERS) + i * BLK + threadIdx.x;
            const long long pix = (long long)b * NPIX + (long long)g * 4;
            v4f gv = *(const v4f*)(gray + pix);
            histo4(gv, mn, scale, sh);
        }
#endif
    } else {
        #pragma unroll
        for (int i = 0; i < HIST_ITERS; ++i) {
            const int g = blockIdx.x * (BLK * HIST_ITERS) + i * BLK + threadIdx.x;
            const long long pix = (long long)b * NPIX + (long long)g * 4;
            v4f gv = gray4_from_x(x, pix);
            histo4(gv, mn, scale, sh);
        }
    }

    __syncthreads();
    unsigned c = sh[threadIdx.x];
    if (c) atomicAdd(&hist[b * NBINS + threadIdx.x], c);
}

// ---------------- pass 3: Otsu scan (one block per image) ----------------
__global__ __launch_bounds__(NBINS) void otsu_scan_k(const unsigned* __restrict__ hist,
                                                     const unsigned* __restrict__ minmax,
                                                     float* __restrict__ thr) {
    const int b = blockIdx.x;
    const int t = threadIdx.x;
    __shared__ float sHO[NBINS], sHCO[NBINS];   // originals
    __shared__ float sH[NBINS], sHC[NBINS];     // inclusive scans
    __shared__ float rv[NBINS];
    __shared__ int   ri[NBINS];

    const float mn  = __uint_as_float(minmax[2 * b + 0]);
    const float mx  = __uint_as_float(minmax[2 * b + 1]);
    const float rng = fmaxf(mx - mn, 1e-12f);

    const float h      = (float)hist[b * NBINS + t];
    const float center = mn + ((float)t + 0.5f) * (rng / 256.0f);
    const float hc     = h * center;
    sHO[t] = h; sHCO[t] = hc; sH[t] = h; sHC[t] = hc;
    __syncthreads();

    // Hillis-Steele inclusive scans of hist and hist*center
    float a = h, c = hc;
    #pragma unroll
    for (int off = 1; off < NBINS; off <<= 1) {
        float av = 0.0f, cv = 0.0f;
        if (t >= off) { av = sH[t - off]; cv = sHC[t - off]; }
        __syncthreads();
        a += av; c += cv;
        sH[t] = a; sHC[t] = c;
        __syncthreads();
    }

    const float total  = sH[NBINS - 1];
    const float totalC = sHC[NBINS - 1];

    float var12 = -1.0f;
    if (t < NBINS - 1) {
        const float w1 = sH[t];                                   // sum_{j<=t} h
        const float w2 = total - sH[t + 1] + sHO[t + 1];          // sum_{j>=t+1} h
        const float m1 = sHC[t] / fmaxf(w1, 1e-12f);
        const float m2 = (totalC - sHC[t + 1] + sHCO[t + 1]) / fmaxf(w2, 1e-12f);
        const float d  = m1 - m2;
        var12 = w1 * w2 * d * d;
    }
    rv[t] = var12; ri[t] = t;
    __syncthreads();
    // argmax, first index wins ties (matches jnp.argmax)
    for (int s = NBINS / 2; s > 0; s >>= 1) {
        if (t < s) {
            float o = rv[t + s]; int oi = ri[t + s];
            if (o > rv[t] || (o == rv[t] && oi < ri[t])) { rv[t] = o; ri[t] = oi; }
        }
        __syncthreads();
    }
    if (t == 0) thr[b] = mn + ((float)ri[0] + 0.5f) * (rng / 256.0f);
}

// ---------------- pass 4: binarize + broadcast to 3 channels ----------------
__global__ __launch_bounds__(BLK) void otsu_binarize_k(const float* __restrict__ x,
                                                       const float* __restrict__ gray,
                                                       const float* __restrict__ thr,
                                                       float* __restrict__ out) {
    const int b = blockIdx.y;
    const int g = blockIdx.x * BLK + threadIdx.x;
    const long long pix = (long long)b * NPIX + (long long)g * 4;
    const float tv = thr[b];                                       // uniform -> scalar load
    v4f gv = gray ? *(const v4f*)(gray + pix) : gray4_from_x(x, pix);
    const float p0 = gv.x > tv ? 255.0f : 0.0f;
    const float p1 = gv.y > tv ? 255.0f : 0.0f;
    const float p2 = gv.z > tv ? 255.0f : 0.0f;
    const float p3 = gv.w > tv ? 255.0f : 0.0f;
    v4f o0; o0.x = p0; o0.y = p0; o0.z = p0; o0.w = p1;
    v4f o1; o1.x = p1; o1.y = p1; o1.z = p2; o1.w = p2;
    v4f o2; o2.x = p2; o2.y = p3; o2.z = p3; o2.w = p3;
    v4f* ov = (v4f*)(out + pix * 3);
    __builtin_nontemporal_store(o0, ov + 0);                       // 201MB stream-once: NT
    __builtin_nontemporal_store(o1, ov + 1);
    __builtin_nontemporal_store(o2, ov + 2);
}

// ---------------- host launch ----------------
extern "C" void kernel_launch(void* const* d_in, const int* in_sizes, int n_in,
                              void* d_out, int out_size, void* d_ws, size_t ws_size,
                              hipStream_t stream) {
    const float* x  = (const float*)d_in[0];
    float* out      = (float*)d_out;
    char* ws        = (char*)d_ws;

    // ws layout: hist[64*256] u32 | minmax[64*2] u32 | thr[64] f32 | (pad) | gray[64*512*512] f32
    unsigned* hist   = (unsigned*)ws;                              // 65536 B
    unsigned* minmax = (unsigned*)(ws + NIMG * NBINS * 4);         // 512 B
    float*    thr    = (float*)(ws + NIMG * NBINS * 4 + NIMG * 8); // 256 B
    const size_t grayOff   = 66560;                                // 256B-aligned
    const size_t grayBytes = (size_t)NIMG * NPIX * sizeof(float);  // 64 MiB
    float* gray = (ws_size >= grayOff + grayBytes) ? (float*)(ws + grayOff) : nullptr;

    otsu_init_k<<<(NIMG * NBINS + 2 * NIMG + BLK - 1) / BLK, BLK, 0, stream>>>(hist, minmax);
    otsu_gray_minmax_k<<<dim3(GROUPS_PER_IMG / BLK, NIMG), BLK, 0, stream>>>(x, gray, minmax);
    otsu_hist_k<<<dim3(GROUPS_PER_IMG / (BLK * HIST_ITERS), NIMG), BLK, 0, stream>>>(x, gray, minmax, hist);
    otsu_scan_k<<<NIMG, NBINS, 0, stream>>>(hist, minmax, thr);
    otsu_binarize_k<<<dim3(GROUPS_PER_IMG / BLK, NIMG), BLK, 0, stream>>>(x, gray, thr, out);
}